// _EdgeAwareBlock_38319698215654
// MI455X (gfx1250) — compile-verified
//
#include <hip/hip_runtime.h>

#define H 128
#define Nn 1024
#define SCR_STRIDE 1025

typedef __attribute__((ext_vector_type(2))) float v2f;
typedef __attribute__((ext_vector_type(8))) float v8f;
typedef __attribute__((ext_vector_type(4))) int v4i;

#if defined(__has_builtin)
#if __has_builtin(__builtin_amdgcn_global_load_async_to_lds_b128) && \
    __has_builtin(__builtin_amdgcn_s_wait_asynccnt)
#define USE_ASYNC_LDS 1
#endif
#endif
#ifndef USE_ASYNC_LDS
#define USE_ASYNC_LDS 0
#endif

__device__ __forceinline__ v8f wmma4(v2f a, v2f b, v8f c) {
  return __builtin_amdgcn_wmma_f32_16x16x4_f32(false, a, false, b, (short)0, c,
                                               false, false);
}

__device__ __forceinline__ float warp_reduce_max(float v) {
#pragma unroll
  for (int off = 16; off > 0; off >>= 1) v = fmaxf(v, __shfl_xor(v, off, 32));
  return v;
}
__device__ __forceinline__ float warp_reduce_sum(float v) {
#pragma unroll
  for (int off = 16; off > 0; off >>= 1) v += __shfl_xor(v, off, 32);
  return v;
}

__device__ __forceinline__ float silu(float x) {
  return x / (1.0f + __expf(-x));
}

#if USE_ASYNC_LDS
__device__ __forceinline__ void async_b128(const float* g, float* l) {
  __builtin_amdgcn_global_load_async_to_lds_b128(
      (__attribute__((address_space(1))) v4i*)g,
      (__attribute__((address_space(3))) v4i*)l, 0, 0);
}
#endif

// Stage a 16-row x 128-col fp32 tile from global (row stride H) into LDS
// (row stride 132, conflict-free for WMMA A-fragment column reads).
__device__ __forceinline__ void stage_tile(const float* __restrict__ g,
                                           float* l, int tid) {
#if USE_ASYNC_LDS
  // 16B async chunks: 512 chunks, 256 threads -> 2 each; LDS written directly
  // by the async unit, tracked on ASYNCcnt (no VGPR round-trip).
#pragma unroll
  for (int chunk = tid; chunk < 512; chunk += 256) {
    int m = chunk >> 5, c4 = (chunk & 31) * 4;
    async_b128(&g[m * H + c4], &l[m * 132 + c4]);
  }
  __builtin_amdgcn_s_wait_asynccnt(0);
#else
  for (int idx = tid; idx < 16 * H; idx += 256) {
    int m = idx >> 7, c2 = idx & 127;
    l[m * 132 + c2] = g[m * H + c2];
  }
#endif
}

// y[m][n] = sum_k xs[m][k]*W[n][k] + bvec[n], 16 rows x 16 cols tile (nb base)
__device__ __forceinline__ v8f proj_tile(const float* xs /*16x132 LDS*/,
                                         const float* __restrict__ W,
                                         const float* __restrict__ bvec,
                                         int nb, int hl, int r, int ldw) {
  v8f c = {};
  float bias = bvec[nb + r];
#pragma unroll
  for (int g = 0; g < 8; ++g) c[g] = bias;
  for (int kb = 0; kb < H; kb += 4) {
    v2f a, b;
    a.x = xs[r * 132 + kb + 2 * hl];
    a.y = xs[r * 132 + kb + 2 * hl + 1];
    b = *(const v2f*)&W[(nb + r) * ldw + kb + 2 * hl];
    c = wmma4(a, b, c);
  }
  return c;
}

// ---------------- Kernel 1: Q/K/V projections ----------------
__global__ __launch_bounds__(256) void qkv_kernel(
    const float* __restrict__ x, const float* __restrict__ Wq,
    const float* __restrict__ bq, const float* __restrict__ Wk,
    const float* __restrict__ bk, const float* __restrict__ Wv,
    const float* __restrict__ bv, float* __restrict__ q, float* __restrict__ k,
    float* __restrict__ v) {
  __shared__ float xs[16 * 132];
  const int row0 = blockIdx.x * 16;
  const int tid = threadIdx.x;
  stage_tile(&x[row0 * H], xs, tid);
  __syncthreads();
  const int wave = tid >> 5, lane = tid & 31;
  const int hl = lane >> 4, r = lane & 15;
  const int nb = wave * 16;

  v8f cq = proj_tile(xs, Wq, bq, nb, hl, r, H);
  v8f ck = proj_tile(xs, Wk, bk, nb, hl, r, H);
  v8f cv = proj_tile(xs, Wv, bv, nb, hl, r, H);
#pragma unroll
  for (int g = 0; g < 8; ++g) {
    int m = row0 + g + 8 * hl;
    q[m * H + nb + r] = cq[g];
    k[m * H + nb + r] = ck[g];
    v[m * H + nb + r] = cv[g];
  }
}

// ---------------- Kernel 2: edge MLP + attention ----------------
__global__ __launch_bounds__(256) void edge_attn_kernel(
    const float* __restrict__ q, const float* __restrict__ k,
    const float* __restrict__ v, const float* __restrict__ offsets,
    const float* __restrict__ node_mask, const float* __restrict__ We1,
    const float* __restrict__ be1, const float* __restrict__ We2,
    const float* __restrict__ be2, float* __restrict__ update) {
  extern __shared__ float smem[];
  float* qs = smem;                    // 16*132
  float* sc = qs + 16 * 132;           // 16*SCR_STRIDE
  float* offi = sc + 16 * SCR_STRIDE;  // 16*4
  float* mi = offi + 64;               // 16
  float* w1 = mi + 16;                 // 128*4
  float* b1 = w1 + 512;                // 128
  float* w2 = b1 + 128;                // 128

  const int b = blockIdx.x >> 6;
  const int it = blockIdx.x & 63;
  const int row0 = b * Nn + it * 16;
  const int tid = threadIdx.x;

  stage_tile(&q[row0 * H], qs, tid);
  for (int idx = tid; idx < 128; idx += 256) {
    w1[idx * 4 + 0] = We1[idx * 3 + 0];
    w1[idx * 4 + 1] = We1[idx * 3 + 1];
    w1[idx * 4 + 2] = We1[idx * 3 + 2];
    b1[idx] = be1[idx];
    w2[idx] = We2[idx];
  }
  if (tid < 16) {
    offi[tid * 4 + 0] = offsets[(row0 + tid) * 3 + 0];
    offi[tid * 4 + 1] = offsets[(row0 + tid) * 3 + 1];
    offi[tid * 4 + 2] = offsets[(row0 + tid) * 3 + 2];
    mi[tid] = node_mask[row0 + tid];
  }
  __syncthreads();

  const int wave = tid >> 5, lane = tid & 31;
  const int hl = lane >> 4, r = lane & 15;
  const float be2v = be2[0];
  const float scale = 0.08838834764831843f;  // 1/sqrt(128)

  for (int jt = wave; jt < 64; jt += 8) {
    const int j0 = b * Nn + jt * 16;
    // prefetch next j-tile of K into the cache hierarchy
    const int jn = (jt + 8 < 64) ? (b * Nn + (jt + 8) * 16) : j0;
    __builtin_prefetch(&k[jn * H + lane * 64], 0, 1);

    // scores tile via WMMA: q[16xH] . k^T
    v8f c = {};
    for (int kb = 0; kb < H; kb += 4) {
      v2f a, bb;
      a.x = qs[r * 132 + kb + 2 * hl];
      a.y = qs[r * 132 + kb + 2 * hl + 1];
      bb = *(const v2f*)&k[(j0 + r) * H + kb + 2 * hl];
      c = wmma4(a, bb, c);
    }
    // edge-MLP bias for this lane's 8 C-fragment pairs (m = g+8*hl, n = r)
    float ojx = offsets[(j0 + r) * 3 + 0];
    float ojy = offsets[(j0 + r) * 3 + 1];
    float ojz = offsets[(j0 + r) * 3 + 2];
    float mj = node_mask[j0 + r];
    float dx[8], dy[8], dz[8], bias[8];
#pragma unroll
    for (int g = 0; g < 8; ++g) {
      int m = g + 8 * hl;
      dx[g] = offi[m * 4 + 0] - ojx;
      dy[g] = offi[m * 4 + 1] - ojy;
      dz[g] = offi[m * 4 + 2] - ojz;
      bias[g] = be2v;
    }
    for (int h = 0; h < 128; ++h) {
      float wx = w1[h * 4 + 0], wy = w1[h * 4 + 1], wz = w1[h * 4 + 2];
      float bb1 = b1[h], ww2 = w2[h];
#pragma unroll
      for (int g = 0; g < 8; ++g) {
        float t = fmaf(dx[g], wx, fmaf(dy[g], wy, fmaf(dz[g], wz, bb1)));
        bias[g] = fmaf(silu(t), ww2, bias[g]);
      }
    }
#pragma unroll
    for (int g = 0; g < 8; ++g) {
      int m = g + 8 * hl;
      float d2 = dx[g] * dx[g] + dy[g] * dy[g] + dz[g] * dz[g];
      bool live = (d2 <= 9.0f + 1e-6f) && (mi[m] * mj > 0.0f);
      float s = c[g] * scale + bias[g];
      sc[m * SCR_STRIDE + jt * 16 + r] = live ? s : -1e9f;
    }
  }
  __syncthreads();

  // softmax rows (each wave: 2 rows), masked entries -> exact 0, clip sum
  for (int rr = wave * 2; rr < wave * 2 + 2; ++rr) {
    float mx = -3.0e38f;
    for (int jj = lane; jj < Nn; jj += 32)
      mx = fmaxf(mx, sc[rr * SCR_STRIDE + jj]);
    mx = warp_reduce_max(mx);
    float sum = 0.0f;
    for (int jj = lane; jj < Nn; jj += 32) {
      float sv = sc[rr * SCR_STRIDE + jj];
      float e = (sv <= -1e8f) ? 0.0f : __expf(sv - mx);
      sc[rr * SCR_STRIDE + jj] = e;
      sum += e;
    }
    sum = warp_reduce_sum(sum);
    float inv = 1.0f / fmaxf(sum, 1e-6f);
    for (int jj = lane; jj < Nn; jj += 32) sc[rr * SCR_STRIDE + jj] *= inv;
  }
  __syncthreads();

  // update = attn @ v (wave owns one 16-col tile of H)
  {
    const int nb = wave * 16;
    const int vb = b * Nn;
    v8f c = {};
    for (int kb = 0; kb < Nn; kb += 4) {
      v2f a, bb;
      a.x = sc[r * SCR_STRIDE + kb + 2 * hl];
      a.y = sc[r * SCR_STRIDE + kb + 2 * hl + 1];
      bb.x = v[(vb + kb + 2 * hl) * H + nb + r];
      bb.y = v[(vb + kb + 2 * hl + 1) * H + nb + r];
      c = wmma4(a, bb, c);
    }
#pragma unroll
    for (int g = 0; g < 8; ++g)
      update[(row0 + g + 8 * hl) * H + nb + r] = c[g];
  }
}

// ---------------- Kernel 3: output projection + residual + LN1 ----------------
__global__ __launch_bounds__(256) void out_ln_kernel(
    const float* __restrict__ upd, const float* __restrict__ node_h,
    const float* __restrict__ Wo, const float* __restrict__ bo,
    const float* __restrict__ g1, const float* __restrict__ beta1,
    float* __restrict__ h1) {
  __shared__ float us[16 * 132];
  __shared__ float hb[16][128];
  const int row0 = blockIdx.x * 16;
  const int tid = threadIdx.x;
  stage_tile(&upd[row0 * H], us, tid);
  __syncthreads();
  const int wave = tid >> 5, lane = tid & 31;
  const int hl = lane >> 4, r = lane & 15;
  const int nb = wave * 16;

  v8f c = proj_tile(us, Wo, bo, nb, hl, r, H);
#pragma unroll
  for (int g = 0; g < 8; ++g) {
    int m = g + 8 * hl;
    hb[m][nb + r] = c[g] + node_h[(row0 + m) * H + nb + r];
  }
  __syncthreads();
  for (int rr = wave * 2; rr < wave * 2 + 2; ++rr) {
    float s = 0.f, sq = 0.f;
    for (int c2 = lane; c2 < H; c2 += 32) {
      float x = hb[rr][c2];
      s += x;
      sq += x * x;
    }
    s = warp_reduce_sum(s);
    sq = warp_reduce_sum(sq);
    float mean = s * (1.0f / 128.0f);
    float var = sq * (1.0f / 128.0f) - mean * mean;
    float inv = rsqrtf(var + 1e-5f);
    for (int c2 = lane; c2 < H; c2 += 32)
      h1[(row0 + rr) * H + c2] = (hb[rr][c2] - mean) * inv * g1[c2] + beta1[c2];
  }
}

// ---------------- Kernel 4: FFN + residual + LN2 + mask ----------------
__global__ __launch_bounds__(256) void ffn_kernel(
    const float* __restrict__ h1g, const float* __restrict__ Wf1,
    const float* __restrict__ bf1, const float* __restrict__ Wf2,
    const float* __restrict__ bf2, const float* __restrict__ g2,
    const float* __restrict__ beta2, const float* __restrict__ node_mask,
    float* __restrict__ out) {
  __shared__ float hs[16 * 132];
  __shared__ float ts[16 * 260];
  __shared__ float ob[16][128];
  const int row0 = blockIdx.x * 16;
  const int tid = threadIdx.x;
  stage_tile(&h1g[row0 * H], hs, tid);
  __syncthreads();
  const int wave = tid >> 5, lane = tid & 31;
  const int hl = lane >> 4, r = lane & 15;

  // GEMM1: [16,128] @ Wf1^T -> [16,256], SiLU, stage in LDS
  for (int nt = wave; nt < 16; nt += 8) {
    const int nb = nt * 16;
    v8f c = proj_tile(hs, Wf1, bf1, nb, hl, r, H);
#pragma unroll
    for (int g = 0; g < 8; ++g) {
      int m = g + 8 * hl;
      ts[m * 260 + nb + r] = silu(c[g]);
    }
  }
  __syncthreads();

  // GEMM2: [16,256] @ Wf2^T -> [16,128], + residual
  {
    const int nb = wave * 16;
    v8f c = {};
    float bias = bf2[nb + r];
#pragma unroll
    for (int g = 0; g < 8; ++g) c[g] = bias;
    for (int kb = 0; kb < 256; kb += 4) {
      v2f a, bb;
      a.x = ts[r * 260 + kb + 2 * hl];
      a.y = ts[r * 260 + kb + 2 * hl + 1];
      bb = *(const v2f*)&Wf2[(nb + r) * 256 + kb + 2 * hl];
      c = wmma4(a, bb, c);
    }
#pragma unroll
    for (int g = 0; g < 8; ++g) {
      int m = g + 8 * hl;
      ob[m][nb + r] = c[g] + hs[m * 132 + nb + r];
    }
  }
  __syncthreads();

  for (int rr = wave * 2; rr < wave * 2 + 2; ++rr) {
    float s = 0.f, sq = 0.f;
    for (int c2 = lane; c2 < H; c2 += 32) {
      float x = ob[rr][c2];
      s += x;
      sq += x * x;
    }
    s = warp_reduce_sum(s);
    sq = warp_reduce_sum(sq);
    float mean = s * (1.0f / 128.0f);
    float var = sq * (1.0f / 128.0f) - mean * mean;
    float inv = rsqrtf(var + 1e-5f);
    float nm = node_mask[row0 + rr];
    for (int c2 = lane; c2 < H; c2 += 32)
      out[(row0 + rr) * H + c2] =
          ((ob[rr][c2] - mean) * inv * g2[c2] + beta2[c2]) * nm;
  }
}

extern "C" void kernel_launch(void* const* d_in, const int* in_sizes, int n_in,
                              void* d_out, int out_size, void* d_ws,
                              size_t ws_size, hipStream_t stream) {
  const float* node_h = (const float*)d_in[0];
  const float* offsets = (const float*)d_in[1];
  const float* node_mask = (const float*)d_in[2];
  const float* Wq = (const float*)d_in[3];
  const float* bq = (const float*)d_in[4];
  const float* Wk = (const float*)d_in[5];
  const float* bk = (const float*)d_in[6];
  const float* Wv = (const float*)d_in[7];
  const float* bv = (const float*)d_in[8];
  const float* We1 = (const float*)d_in[9];
  const float* be1 = (const float*)d_in[10];
  const float* We2 = (const float*)d_in[11];
  const float* be2 = (const float*)d_in[12];
  const float* Wo = (const float*)d_in[13];
  const float* bo = (const float*)d_in[14];
  const float* g1 = (const float*)d_in[15];
  const float* beta1 = (const float*)d_in[16];
  const float* Wf1 = (const float*)d_in[17];
  const float* bf1 = (const float*)d_in[18];
  const float* Wf2 = (const float*)d_in[19];
  const float* bf2 = (const float*)d_in[20];
  const float* g2 = (const float*)d_in[21];
  const float* beta2 = (const float*)d_in[22];

  const int ntok = 2 * Nn;  // 2048
  float* q = (float*)d_ws;
  float* k = q + ntok * H;
  float* v = k + ntok * H;
  float* upd = v + ntok * H;
  float* h1 = upd + ntok * H;

  const int blocks = ntok / 16;  // 128
  qkv_kernel<<<blocks, 256, 0, stream>>>(node_h, Wq, bq, Wk, bk, Wv, bv, q, k,
                                         v);
  const size_t smem2 =
      (16 * 132 + 16 * SCR_STRIDE + 64 + 16 + 512 + 128 + 128) * sizeof(float);
  edge_attn_kernel<<<blocks, 256, smem2, stream>>>(
      q, k, v, offsets, node_mask, We1, be1, We2, be2, upd);
  out_ln_kernel<<<blocks, 256, 0, stream>>>(upd, node_h, Wo, bo, g1, beta1, h1);
  ffn_kernel<<<blocks, 256, 0, stream>>>(h1, Wf1, bf1, Wf2, bf2, g2, beta2,
                                         node_mask, (float*)d_out);
}